// SymmetryLoss_9758165696606
// MI455X (gfx1250) — compile-verified
//
#include <hip/hip_runtime.h>
#include <hip/hip_bf16.h>

typedef __attribute__((ext_vector_type(2))) float v2f;
typedef __attribute__((ext_vector_type(8))) float v8f;

constexpr int kN             = 4096;
constexpr int kB             = 16;
constexpr int kWaves         = 8;
constexpr int kThreads       = kWaves * 32;          // 256
constexpr int kRowsPerBlock  = kWaves * 16;          // 128 rows per block (16 per wave)
constexpr int kBlocksPerBatch= kN / kRowsPerBlock;   // 32
constexpr int kBlocks        = kB * kBlocksPerBatch; // 512
constexpr int kColTiles      = kN / 16;              // 256

#if defined(__has_builtin)
#  if __has_builtin(__builtin_amdgcn_wmma_f32_16x16x4_f32)
#    define USE_WMMA_F32X4 1
#  endif
#endif

// One block per (batch, 128-row tile). Each wave owns a 16-row WMMA tile and
// scans all 4096 columns from LDS, keeping a per-lane running min of d2.
__global__ __launch_bounds__(kThreads)
void sym_chamfer_rowmin(const float* __restrict__ xyz, float* __restrict__ partial) {
  __shared__ float4 pts[kN];   // (x, y, z, |p|^2)  -> exactly 64 KB

  const int batch = blockIdx.x / kBlocksPerBatch;
  const int rtile = blockIdx.x % kBlocksPerBatch;
  const float* bx = xyz + (size_t)batch * 3u * kN;   // (B,3,N) planar layout

  // Cooperative, coalesced stage of this batch's points into LDS.
  for (int i = threadIdx.x; i < kN; i += kThreads) {
    float x = bx[i];
    float y = bx[kN + i];
    float z = bx[2 * kN + i];
    pts[i] = make_float4(x, y, z, x * x + y * y + z * z);
  }
  __syncthreads();

  const int  lane = threadIdx.x & 31;
  const int  wave = threadIdx.x >> 5;
  const int  l    = lane & 15;
  const bool hi   = lane >= 16;
  const int  rb   = rtile * kRowsPerBlock + wave * 16;

  // |a_row|^2 term per C/D VGPR r: rows rb+r (lanes 0-15) / rb+8+r (lanes 16-31).
  // Mirror is an isometry, so |a_i|^2 == |p_i|^2 (precomputed in .w).
  float a2r[8];
  #pragma unroll
  for (int r = 0; r < 8; ++r) a2r[r] = pts[rb + r + (hi ? 8 : 0)].w;

#if USE_WMMA_F32X4
  // A operand (16x4, v2f per lane), pre-scaled by -2 with the yz-mirror folded in:
  // A'[i][:] = -2 * (-x_i, y_i, z_i, 0) = (2x_i, -2y_i, -2z_i, 0)
  // lanes 0-15: {K0,K1} = {2x, -2y}; lanes 16-31: {K2,K3} = {-2z, 0}
  float4 ap = pts[rb + l];
  v2f aop;
  aop.x = hi ? (-2.0f * ap.z) : (2.0f * ap.x);
  aop.y = hi ? 0.0f           : (-2.0f * ap.y);
#else
  float3 arow[8];
  #pragma unroll
  for (int r = 0; r < 8; ++r) {
    float4 p = pts[rb + r + (hi ? 8 : 0)];
    arow[r] = make_float3(-p.x, p.y, p.z);   // mirrored point
  }
#endif

  v8f runmin;
  #pragma unroll
  for (int r = 0; r < 8; ++r) runmin[r] = __builtin_inff();

  for (int ct = 0; ct < kColTiles; ++ct) {
    float4 bp = pts[ct * 16 + l];            // ds_load_b128: column point + |b|^2
#if USE_WMMA_F32X4
    // B operand (4x16): lanes 0-15 {K0,K1} = {x, y}; lanes 16-31 {K2,K3} = {z, 0}
    v2f bop;
    bop.x = hi ? bp.z : bp.x;
    bop.y = hi ? 0.0f : bp.y;
    // C[i][j] = |a_i|^2 + |b_j|^2  ->  D = A'*B + C = d2 tile exactly (f32)
    v8f c;
    #pragma unroll
    for (int r = 0; r < 8; ++r) c[r] = a2r[r] + bp.w;
    v8f d = __builtin_amdgcn_wmma_f32_16x16x4_f32(false, aop, false, bop,
                                                  (short)0, c, false, false);
    #pragma unroll
    for (int r = 0; r < 8; ++r) runmin[r] = fminf(runmin[r], d[r]);
#else
    #pragma unroll
    for (int r = 0; r < 8; ++r) {
      float dot = arow[r].x * bp.x + arow[r].y * bp.y + arow[r].z * bp.z;
      float d2  = a2r[r] + bp.w - 2.0f * dot;
      runmin[r] = fminf(runmin[r], d2);
    }
#endif
  }

  // Row-min: reduce across each 16-lane half (xor masks < 16 stay in-half).
  #pragma unroll
  for (int r = 0; r < 8; ++r) {
    float v = runmin[r];
    v = fminf(v, __shfl_xor(v, 8, 32));
    v = fminf(v, __shfl_xor(v, 4, 32));
    v = fminf(v, __shfl_xor(v, 2, 32));
    v = fminf(v, __shfl_xor(v, 1, 32));
    runmin[r] = v;
  }
  float s = 0.0f;
  #pragma unroll
  for (int r = 0; r < 8; ++r) s += runmin[r];   // lanes<16: rows rb..rb+7; lanes>=16: rb+8..rb+15
  s += __shfl_xor(s, 16, 32);                   // full 16-row sum for this wave

  // Deterministic per-block sum; reuse pts LDS after all waves finished reading.
  __syncthreads();
  float* wsum = (float*)pts;
  if (lane == 0) wsum[wave] = s;
  __syncthreads();
  if (threadIdx.x == 0) {
    float t = 0.0f;
    for (int w = 0; w < kWaves; ++w) t += wsum[w];
    partial[blockIdx.x] = t;
  }
}

// Fixed-order final reduction: loss = 2 * sum(rowmins) / (B*N)
// (dist21 == dist12 because the mirror is an isometric involution).
__global__ void sym_chamfer_finalize(const float* __restrict__ partial,
                                     float* __restrict__ out) {
  if (threadIdx.x == 0 && blockIdx.x == 0) {
    float s = 0.0f;
    for (int i = 0; i < kBlocks; ++i) s += partial[i];
    out[0] = 2.0f * s / (float)(kB * kN);
  }
}

extern "C" void kernel_launch(void* const* d_in, const int* in_sizes, int n_in,
                              void* d_out, int out_size, void* d_ws, size_t ws_size,
                              hipStream_t stream) {
  (void)in_sizes; (void)n_in; (void)out_size; (void)ws_size;
  const float* xyz    = (const float*)d_in[0];
  float*       out    = (float*)d_out;
  float*       partial= (float*)d_ws;    // 512 floats = 2 KB scratch

  sym_chamfer_rowmin <<<kBlocks, kThreads, 0, stream>>>(xyz, partial);
  sym_chamfer_finalize<<<1, 32, 0, stream>>>(partial, out);
}